// RRWPStemV2EdgeEncoder_66271345377493
// MI455X (gfx1250) — compile-verified
//
#include <hip/hip_runtime.h>

// Problem constants (match reference)
#define B_GRAPHS 64
#define N_NODES  128
#define DEG      16
#define EMB      24
#define OUTC     64
#define EPSN     1e-5f
#define M_ROWS     (B_GRAPHS * N_NODES)              // 8192 nodes total
#define TOTAL_ROWS (B_GRAPHS * N_NODES * N_NODES)    // 1,048,576 output rows
#define E_EDGES    (B_GRAPHS * N_NODES * DEG)        // 131,072 edges

typedef float v2f __attribute__((ext_vector_type(2)));
typedef float v8f __attribute__((ext_vector_type(8)));

// ---------------------------------------------------------------------------
// Kernel 1: both projections rwse @ w_src and rwse @ w_dst via fp32 WMMA.
// One wave32 owns one 16(M)x16(N) output tile; K=24 = 6 steps of
// V_WMMA_F32_16X16X4_F32 (exact fp32, no precision loss vs reference).
// A 16x4 layout: lanes 0-15 hold M=lane {K=k0,k0+1}, lanes 16-31 {K=k0+2,k0+3}.
// B 4x16 mirrored; C/D: VGPR r -> M = r + 8*(lane>=16), N = lane&15.
// ---------------------------------------------------------------------------
__global__ void k_gemm_wmma(const float* __restrict__ rwse,
                            const float* __restrict__ w_src,
                            const float* __restrict__ w_dst,
                            float* __restrict__ feats /* [2][8192][64] */) {
  const int wave = (blockIdx.x * blockDim.x + threadIdx.x) >> 5;  // 0..4095
  const int lane = threadIdx.x & 31;
  const int s    = wave & 1;          // 0 = src weights, 1 = dst weights
  const int nt   = (wave >> 1) & 3;   // N tile (64/16)
  const int mt   = wave >> 3;         // M tile (8192/16)
  const float* __restrict__ W = s ? w_dst : w_src;
  float* __restrict__ F = feats + (size_t)s * (M_ROWS * OUTC);

  const int half = lane >> 4;
  const int l16  = lane & 15;
  const int mrow = mt * 16 + l16;     // A-matrix row for this lane
  const int ncol = nt * 16 + l16;     // B/C column for this lane

  v8f acc = {};
#pragma unroll
  for (int k0 = 0; k0 < EMB; k0 += 4) {
    const int kb = k0 + 2 * half;
    v2f a, b;
    a.x = rwse[mrow * EMB + kb];
    a.y = rwse[mrow * EMB + kb + 1];
    b.x = W[kb * OUTC + ncol];
    b.y = W[(kb + 1) * OUTC + ncol];
    acc = __builtin_amdgcn_wmma_f32_16x16x4_f32(false, a, false, b,
                                                (short)0, acc, false, false);
  }
#pragma unroll
  for (int r = 0; r < 8; ++r) {
    const int m = mt * 16 + half * 8 + r;
    F[m * OUTC + ncol] = acc[r];
  }
}

// ---------------------------------------------------------------------------
// Kernel 2: zero the stats accumulators (sum[64], sumsq[64], + scale/shift).
// ---------------------------------------------------------------------------
__global__ void k_init_stats(float* __restrict__ stats) {
  stats[threadIdx.x] = 0.0f;
}

// ---------------------------------------------------------------------------
// Kernel 3: the 256MB streaming write out[g,u,v,:] = src[g,u,:] + dst[g,v,:].
// 16 lanes per 64-float row, float4 (b128) loads/stores. Feats (4MB) stay in L2.
// ---------------------------------------------------------------------------
__global__ void k_broadcast(const float* __restrict__ feats,
                            float* __restrict__ out) {
  const unsigned t = blockIdx.x * blockDim.x + threadIdx.x;  // 16.7M threads
  const unsigned r = t >> 4;           // output row
  const unsigned q = t & 15;           // float4 slot within row
  const unsigned g = r >> 14;
  const unsigned u = (r >> 7) & (N_NODES - 1);
  const unsigned v = r & (N_NODES - 1);
  const float4 a = ((const float4*)(feats + (size_t)(g * N_NODES + u) * OUTC))[q];
  const float4 b = ((const float4*)(feats + (size_t)(M_ROWS + g * N_NODES + v) * OUTC))[q];
  const float4 o = make_float4(a.x + b.x, a.y + b.y, a.z + b.z, a.w + b.w);
  ((float4*)out)[(size_t)r * 16 + q] = o;
}

// ---------------------------------------------------------------------------
// Kernel 4: analytic per-channel sum/sumsq of the dense (s+d) tensor, computed
// from the 4MB feats instead of re-reading 256MB:
//   sum_g   = N*(Σ_u s + Σ_v d)
//   sumsq_g = N*Σ s^2 + 2*(Σ s)(Σ d) + N*Σ d^2
// Grid: one block per graph, one thread per channel.
// ---------------------------------------------------------------------------
__global__ void k_stats_feats(const float* __restrict__ feats,
                              float* __restrict__ stats) {
  const int g = blockIdx.x;
  const int c = threadIdx.x;
  const float* __restrict__ sp = feats + (size_t)(g * N_NODES) * OUTC + c;
  const float* __restrict__ dp = sp + (size_t)M_ROWS * OUTC;
  float Ss = 0.f, Ss2 = 0.f, Sd = 0.f, Sd2 = 0.f;
#pragma unroll 4
  for (int u = 0; u < N_NODES; ++u) {
    const float sv = sp[u * OUTC];
    const float dv = dp[u * OUTC];
    Ss += sv; Ss2 += sv * sv;
    Sd += dv; Sd2 += dv * dv;
  }
  unsafeAtomicAdd(&stats[c], (float)N_NODES * (Ss + Sd));
  unsafeAtomicAdd(&stats[64 + c],
                  (float)N_NODES * Ss2 + 2.0f * Ss * Sd + (float)N_NODES * Sd2);
}

// ---------------------------------------------------------------------------
// Kernel 5: scatter-add edge_attr into out with native f32 atomics.
// flat = edge_index[0]*N + (edge_index[1] % N). The atomicAdd RETURN value
// yields the exact sumsq correction 2*old*e + e^2, which telescopes to
// final^2 - initial^2 per cell even when duplicate edges collide.
// Block: 256 threads = 16 edges x 16 float4 channel-groups, 16 iterations
// (256 edges/block), LDS-reduced so only 128 stat atomics per block.
// ---------------------------------------------------------------------------
__global__ void k_scatter(const int* __restrict__ ei,
                          const float* __restrict__ edge_attr,
                          float* __restrict__ out,
                          float* __restrict__ stats) {
  const int el = threadIdx.x >> 4;   // edge slot 0..15
  const int q  = threadIdx.x & 15;   // float4 channel group
  const int base = blockIdx.x * 256;
  float4 corr = make_float4(0.f, 0.f, 0.f, 0.f);
  float4 esum = make_float4(0.f, 0.f, 0.f, 0.f);
#pragma unroll 1
  for (int it = 0; it < 16; ++it) {
    const int e   = base + it * 16 + el;
    const int src = ei[e];
    const int dst = ei[E_EDGES + e];
    const size_t flat = (size_t)src * N_NODES + (dst & (N_NODES - 1));
    float* __restrict__ p = out + flat * OUTC + q * 4;
    const float4 ev = ((const float4*)(edge_attr + (size_t)e * OUTC))[q];
    float old;
    old = unsafeAtomicAdd(p + 0, ev.x); corr.x += (2.f * old + ev.x) * ev.x; esum.x += ev.x;
    old = unsafeAtomicAdd(p + 1, ev.y); corr.y += (2.f * old + ev.y) * ev.y; esum.y += ev.y;
    old = unsafeAtomicAdd(p + 2, ev.z); corr.z += (2.f * old + ev.z) * ev.z; esum.z += ev.z;
    old = unsafeAtomicAdd(p + 3, ev.w); corr.w += (2.f * old + ev.w) * ev.w; esum.w += ev.w;
  }
  __shared__ float4 lc[256];
  __shared__ float4 le[256];
  lc[threadIdx.x] = corr;
  le[threadIdx.x] = esum;
  __syncthreads();
  if (threadIdx.x < 16) {
    float4 c = make_float4(0.f, 0.f, 0.f, 0.f);
    float4 s = make_float4(0.f, 0.f, 0.f, 0.f);
#pragma unroll
    for (int e = 0; e < 16; ++e) {
      const float4 a = lc[e * 16 + threadIdx.x];
      const float4 b = le[e * 16 + threadIdx.x];
      c.x += a.x; c.y += a.y; c.z += a.z; c.w += a.w;
      s.x += b.x; s.y += b.y; s.z += b.z; s.w += b.w;
    }
    const int cb = threadIdx.x * 4;
    unsafeAtomicAdd(&stats[cb + 0], s.x);
    unsafeAtomicAdd(&stats[cb + 1], s.y);
    unsafeAtomicAdd(&stats[cb + 2], s.z);
    unsafeAtomicAdd(&stats[cb + 3], s.w);
    unsafeAtomicAdd(&stats[64 + cb + 0], c.x);
    unsafeAtomicAdd(&stats[64 + cb + 1], c.y);
    unsafeAtomicAdd(&stats[64 + cb + 2], c.z);
    unsafeAtomicAdd(&stats[64 + cb + 3], c.w);
  }
}

// ---------------------------------------------------------------------------
// Kernel 6: fold stats into per-channel scale/shift:
//   y = x * (gamma * rsqrt(var+eps)) + (beta - mean * that)
// ---------------------------------------------------------------------------
__global__ void k_finalize(const float* __restrict__ gamma,
                           const float* __restrict__ beta,
                           float* __restrict__ stats) {
  const int c = threadIdx.x;
  const float invM = 1.0f / (float)TOTAL_ROWS;
  const float mean = stats[c] * invM;
  const float var  = stats[64 + c] * invM - mean * mean;
  const float sc   = gamma[c] * rsqrtf(var + EPSN);
  stats[128 + c] = sc;
  stats[192 + c] = beta[c] - mean * sc;
}

// ---------------------------------------------------------------------------
// Kernel 7: normalize in place (256MB read + 256MB write, b128 per lane).
// ---------------------------------------------------------------------------
__global__ void k_norm(float* __restrict__ out,
                       const float* __restrict__ stats) {
  const size_t t = (size_t)blockIdx.x * blockDim.x + threadIdx.x;  // float4 idx
  const int q = (int)(t & 15);
  const float4 sc = ((const float4*)(stats + 128))[q];
  const float4 sh = ((const float4*)(stats + 192))[q];
  float4 x = ((float4*)out)[t];
  x.x = x.x * sc.x + sh.x;
  x.y = x.y * sc.y + sh.y;
  x.z = x.z * sc.z + sh.z;
  x.w = x.w * sc.w + sh.w;
  ((float4*)out)[t] = x;
}

extern "C" void kernel_launch(void* const* d_in, const int* in_sizes, int n_in,
                              void* d_out, int out_size, void* d_ws, size_t ws_size,
                              hipStream_t stream) {
  const float* rwse       = (const float*)d_in[0];  // [8192, 24]
  const float* edge_attr  = (const float*)d_in[1];  // [131072, 64]
  const float* w_src      = (const float*)d_in[2];  // [24, 64]
  const float* w_dst      = (const float*)d_in[3];  // [24, 64]
  const float* gamma      = (const float*)d_in[4];  // [64]
  const float* beta       = (const float*)d_in[5];  // [64]
  const int*   edge_index = (const int*)d_in[6];    // [2, 131072]
  float* out = (float*)d_out;                       // [1048576, 64]

  // Workspace: feats [2][8192][64] f32 (4MB) + stats (256 f32)
  float* feats = (float*)d_ws;
  float* stats = feats + (size_t)2 * M_ROWS * OUTC;

  k_init_stats<<<1, 256, 0, stream>>>(stats);
  // 4096 waves = 512 blocks x 8 waves: (512 M-tiles) x (4 N-tiles) x (2 mats)
  k_gemm_wmma<<<512, 256, 0, stream>>>(rwse, w_src, w_dst, feats);
  // 1,048,576 rows x 16 lanes = 16.7M threads
  k_broadcast<<<65536, 256, 0, stream>>>(feats, out);
  k_stats_feats<<<B_GRAPHS, OUTC, 0, stream>>>(feats, stats);
  // 131072 edges / 256 per block
  k_scatter<<<E_EDGES / 256, 256, 0, stream>>>(edge_index, edge_attr, out, stats);
  k_finalize<<<1, OUTC, 0, stream>>>(gamma, beta, stats);
  // 16,777,216 float4s
  k_norm<<<65536, 256, 0, stream>>>(out, stats);
}